// Item_GraphConvolution_mid_attention_over_57698590654951
// MI455X (gfx1250) — compile-verified
//
#include <hip/hip_runtime.h>
#include <hip/hip_bf16.h>

// CDNA5 / gfx1250, wave32. FP32 WMMA (V_WMMA_F32_16X16X4_F32) GEMM pipeline.
// Block = 256 threads = 8 waves. Each block owns 64 rows x 128 cols of output;
// each wave owns 4 row-tiles (16x16) in one 16-col slab -> B fragment is
// reused across 4 WMMAs per k-step.

typedef __attribute__((ext_vector_type(2))) float v2f;
typedef __attribute__((ext_vector_type(8))) float v8f;

#define E_DIM 128
#define F_IN  256
#define RPB   64          // rows per block
#define NT    4           // 16-row tiles per wave

// ---------------------------------------------------------------------------
// 0) zero adj_support accumulator (re-zeroed every call: deterministic)
// ---------------------------------------------------------------------------
__global__ void zero_kernel(float4* __restrict__ p, long long n4) {
    long long i = (long long)blockIdx.x * blockDim.x + threadIdx.x;
    if (i < n4) p[i] = make_float4(0.f, 0.f, 0.f, 0.f);
}

// ---------------------------------------------------------------------------
// 1) fold 5x2 gammas + alpha=linspace(-eps,1+eps,2) into ca, cb
// ---------------------------------------------------------------------------
__global__ void coeff_kernel(const float* __restrict__ lg,
                             const float* __restrict__ mg,
                             float* __restrict__ coeffs) {
    if (threadIdx.x == 0) {
        const float a0 = -1e-9f, a1 = 1.0f + 1e-9f;
        float al = 0.f, bl = 0.f, am = 0.f, cm = 0.f;
        for (int i = 0; i < 5; ++i) {
            float g0 = fmaxf(lg[2*i], 0.f), g1 = fmaxf(lg[2*i+1], 0.f);
            al += g0 * a0 + g1 * a1;
            bl += g0 * (1.f - a0) + g1 * (1.f - a1);
            float m0 = fmaxf(mg[2*i], 0.f), m1 = fmaxf(mg[2*i+1], 0.f);
            am += m0 + m1;
            cm += m0 * a0 + m1 * a1;
        }
        coeffs[0] = al + am;   // ca
        coeffs[1] = bl - cm;   // cb
    }
}

// ---------------------------------------------------------------------------
// 2) support = relu(feature @ weight)   [N,256] x [256,128] -> [N,128]
// ---------------------------------------------------------------------------
__global__ __launch_bounds__(256) void gemm_support_kernel(
        const float* __restrict__ feature, const float* __restrict__ weight,
        float* __restrict__ support, int n) {
    const int lane = threadIdx.x & 31;
    const int wave = threadIdx.x >> 5;
    const int lm = lane & 15;       // M index (A) / N index (B,C)
    const int lh = lane >> 4;       // K-half selector
    const int row0 = blockIdx.x * RPB;
    const int col0 = wave * 16;

    const float* arow[NT];
    #pragma unroll
    for (int t = 0; t < NT; ++t) {
        int ar = row0 + 16 * t + lm;
        if (ar >= n) ar = n - 1;                          // clamp, no divergence
        arow[t] = feature + (long long)ar * F_IN;
    }

    v8f acc[NT];
    #pragma unroll
    for (int t = 0; t < NT; ++t) acc[t] = (v8f){};

    #pragma unroll 2
    for (int k0 = 0; k0 < F_IN; k0 += 4) {
        const int ka = k0 + 2 * lh;
        v2f b;                                            // B[k][n]: n = col0+lm
        b.x = weight[(long long)ka       * E_DIM + (col0 + lm)];
        b.y = weight[(long long)(ka + 1) * E_DIM + (col0 + lm)];
        #pragma unroll
        for (int t = 0; t < NT; ++t) {
            v2f a = *(const v2f*)(arow[t] + ka);          // contiguous b64
            acc[t] = __builtin_amdgcn_wmma_f32_16x16x4_f32(
                         false, a, false, b, (short)0, acc[t], false, false);
        }
    }

    // C layout: VGPR v -> row (16t + v + 8*lh), col = col0+lm ; relu
    if (row0 + RPB <= n) {                                // fast path: no guards
        #pragma unroll
        for (int t = 0; t < NT; ++t) {
            float* base = support + (long long)(row0 + 16*t + 8*lh) * E_DIM + (col0 + lm);
            #pragma unroll
            for (int v = 0; v < 8; ++v)
                base[(long long)v * E_DIM] = fmaxf(acc[t][v], 0.f);
        }
    } else {                                              // tail block only
        for (int t = 0; t < NT; ++t)
            for (int v = 0; v < 8; ++v) {
                int r = row0 + 16*t + v + 8*lh;
                if (r < n)
                    support[(long long)r * E_DIM + (col0 + lm)] = fmaxf(acc[t][v], 0.f);
            }
    }
}

// ---------------------------------------------------------------------------
// 3) adj[row_e] += w_e * support[col_e]  — one wave per edge, float4/lane,
//    hardware FP32 atomics (L2-resident destination, 25.6 MB)
// ---------------------------------------------------------------------------
__global__ __launch_bounds__(256) void scatter_kernel(
        const int* __restrict__ erow, const int* __restrict__ ecol,
        const float* __restrict__ ew, const float* __restrict__ support,
        float* __restrict__ adj, int n_edges) {
    long long tid = (long long)blockIdx.x * blockDim.x + threadIdx.x;
    long long e = tid >> 5;
    if (e >= n_edges) return;
    const int c = (int)(tid & 31) * 4;                    // 32 lanes x 4 = 128 feats
    const int r = erow[e], cl = ecol[e];
    const float w = ew[e];
    const float4 s = *(const float4*)(support + (long long)cl * E_DIM + c);
    float* dst = adj + (long long)r * E_DIM + c;
    unsafeAtomicAdd(dst + 0, w * s.x);
    unsafeAtomicAdd(dst + 1, w * s.y);
    unsafeAtomicAdd(dst + 2, w * s.z);
    unsafeAtomicAdd(dst + 3, w * s.w);
}

// ---------------------------------------------------------------------------
// 4) out = (ca*adj + cb*support) @ v_w.T + 2*v_b
//    (softmax over the query axis makes the attention collapse to v0+v1)
// ---------------------------------------------------------------------------
__global__ __launch_bounds__(256) void gemm_out_kernel(
        const float* __restrict__ support, const float* __restrict__ adj,
        const float* __restrict__ coeffs, const float* __restrict__ v_w,
        const float* __restrict__ v_b, float* __restrict__ out, int n) {
    const int lane = threadIdx.x & 31;
    const int wave = threadIdx.x >> 5;
    const int lm = lane & 15;
    const int lh = lane >> 4;
    const int row0 = blockIdx.x * RPB;
    const int col0 = wave * 16;
    const float ca = coeffs[0], cb = coeffs[1];

    const float* srow[NT];
    const float* grow[NT];
    #pragma unroll
    for (int t = 0; t < NT; ++t) {
        int ar = row0 + 16 * t + lm;
        if (ar >= n) ar = n - 1;
        srow[t] = support + (long long)ar * E_DIM;
        grow[t] = adj     + (long long)ar * E_DIM;
    }
    const float* __restrict__ bv = v_w + (long long)(col0 + lm) * E_DIM;  // B[k][n]=v_w[n][k]

    v8f acc[NT];
    #pragma unroll
    for (int t = 0; t < NT; ++t) acc[t] = (v8f){};

    #pragma unroll 2
    for (int k0 = 0; k0 < E_DIM; k0 += 4) {
        const int ka = k0 + 2 * lh;
        v2f b = *(const v2f*)(bv + ka);                   // contiguous b64
        #pragma unroll
        for (int t = 0; t < NT; ++t) {
            v2f ga = *(const v2f*)(grow[t] + ka);
            v2f sa = *(const v2f*)(srow[t] + ka);
            v2f a;
            a.x = ca * ga.x + cb * sa.x;
            a.y = ca * ga.y + cb * sa.y;
            acc[t] = __builtin_amdgcn_wmma_f32_16x16x4_f32(
                         false, a, false, b, (short)0, acc[t], false, false);
        }
    }

    const float bias2 = 2.0f * v_b[col0 + lm];
    if (row0 + RPB <= n) {
        #pragma unroll
        for (int t = 0; t < NT; ++t) {
            float* base = out + (long long)(row0 + 16*t + 8*lh) * E_DIM + (col0 + lm);
            #pragma unroll
            for (int v = 0; v < 8; ++v)
                base[(long long)v * E_DIM] = acc[t][v] + bias2;
        }
    } else {
        for (int t = 0; t < NT; ++t)
            for (int v = 0; v < 8; ++v) {
                int r = row0 + 16*t + v + 8*lh;
                if (r < n)
                    out[(long long)r * E_DIM + (col0 + lm)] = acc[t][v] + bias2;
            }
    }
}

// ---------------------------------------------------------------------------
extern "C" void kernel_launch(void* const* d_in, const int* in_sizes, int n_in,
                              void* d_out, int out_size, void* d_ws, size_t ws_size,
                              hipStream_t stream) {
    const float* feature   = (const float*)d_in[0];
    const int*   edge_idx  = (const int*)d_in[1];     // [2, E] row-major
    const float* edge_w    = (const float*)d_in[2];
    const float* weight    = (const float*)d_in[3];
    const float* low_gamma = (const float*)d_in[4];
    const float* mid_gamma = (const float*)d_in[5];
    // d_in[6..9] = q_w,q_b,k_w,k_b : mathematically cancelled (query-axis softmax)
    const float* v_w       = (const float*)d_in[10];
    const float* v_b       = (const float*)d_in[11];

    const int n       = in_sizes[0] / F_IN;           // 50000
    const int n_edges = in_sizes[2];                  // 600000

    float* ws      = (float*)d_ws;
    float* coeffs  = ws;                              // 16 floats reserved
    float* support = ws + 16;                         // [n,128]
    float* adj     = support + (long long)n * E_DIM;  // [n,128]

    const int row_blocks = (n + RPB - 1) / RPB;

    // 0) zero adj accumulator
    {
        long long n4 = ((long long)n * E_DIM) / 4;
        int blocks = (int)((n4 + 255) / 256);
        zero_kernel<<<blocks, 256, 0, stream>>>((float4*)adj, n4);
    }
    // 1) fold gammas into (ca, cb)
    coeff_kernel<<<1, 32, 0, stream>>>(low_gamma, mid_gamma, coeffs);
    // 2) support = relu(feature @ weight)
    gemm_support_kernel<<<row_blocks, 256, 0, stream>>>(feature, weight, support, n);
    // 3) scatter-add over edges
    {
        long long threads = (long long)n_edges * 32;
        int blocks = (int)((threads + 255) / 256);
        scatter_kernel<<<blocks, 256, 0, stream>>>(edge_idx, edge_idx + n_edges,
                                                   edge_w, support, adj, n_edges);
    }
    // 4) out = (ca*adj + cb*support) @ v_w.T + 2*v_b
    gemm_out_kernel<<<row_blocks, 256, 0, stream>>>(support, adj, coeffs, v_w, v_b,
                                                    (float*)d_out, n);
}